// LDDMMVariational_31361851195747
// MI455X (gfx1250) — compile-verified
//
#include <hip/hip_runtime.h>

typedef float v2f __attribute__((ext_vector_type(2)));
typedef float v8f __attribute__((ext_vector_type(8)));

#define GAMMA_F 100.0f                       // 1/sigma^2, sigma=0.1
#define C2EXP  (-144.26950408889634f)        // -GAMMA * log2(e)

__global__ __launch_bounds__(256)
void lddmm_evolve_kernel(const float* __restrict__ mom,
                         const float* __restrict__ x,
                         float* __restrict__ out, int N) {
    const int lane = threadIdx.x & 31;
    const int wave = threadIdx.x >> 5;
    const int i0   = blockIdx.x << 4;         // 16 rows per block
    const bool hi  = (lane >= 16);
    const int l15  = lane & 15;

    __shared__ float x2s[16];
    __shared__ float partial[8][16][8];       // [wave][row][quantity 0..6]

    if (threadIdx.x < 16) {
        int i = i0 + threadIdx.x;
        float a = x[3*i], b = x[3*i+1], c = x[3*i+2];
        x2s[threadIdx.x] = a*a + b*b + c*c;
    }
    __syncthreads();

    // A-matrix layout (16x4 f32): lanes 0-15 -> K=0,1 ; lanes 16-31 -> K=2,3(pad 0)
    const int ia = i0 + l15;
    v2f Ax, Ap;
    if (!hi) {
        Ax[0] = x[3*ia];     Ax[1] = x[3*ia+1];
        Ap[0] = mom[3*ia];   Ap[1] = mom[3*ia+1];
    } else {
        Ax[0] = x[3*ia+2];   Ax[1] = 0.0f;
        Ap[0] = mom[3*ia+2]; Ap[1] = 0.0f;
    }

    float x2r[8];
#pragma unroll
    for (int r = 0; r < 8; ++r) x2r[r] = x2s[r + (hi ? 8 : 0)];

    float rs[8], dxa[8], dxb[8], dxc[8], wxa[8], wxb[8], wxc[8];
#pragma unroll
    for (int r = 0; r < 8; ++r) {
        rs[r] = 0.0f;
        dxa[r] = dxb[r] = dxc[r] = 0.0f;
        wxa[r] = wxb[r] = wxc[r] = 0.0f;
    }

    const int ntiles = N >> 4;
    for (int jt = wave; jt < ntiles; jt += 8) {
        const int jj = (jt << 4) + l15;
        float xj0 = x[3*jj],   xj1 = x[3*jj+1],   xj2 = x[3*jj+2];
        float pj0 = mom[3*jj], pj1 = mom[3*jj+1], pj2 = mom[3*jj+2];

        if (jt + 8 < ntiles) {                 // prefetch next j-tile for this wave
            __builtin_prefetch(x   + 3*(jj + 128), 0, 0);
            __builtin_prefetch(mom + 3*(jj + 128), 0, 0);
        }

        // B-matrix layout (4x16 f32): lanes 0-15 -> K=0,1 ; lanes 16-31 -> K=2,3(pad 0)
        v2f Bx, Bp;
        if (!hi) { Bx[0] = xj0; Bx[1] = xj1; Bp[0] = pj0; Bp[1] = pj1; }
        else     { Bx[0] = xj2; Bx[1] = 0.0f; Bp[0] = pj2; Bp[1] = 0.0f; }

        v8f c0 = {0,0,0,0,0,0,0,0};
        // XY[i][j] = <x_i, x_j> ; PP[i][j] = <p_i, p_j>
        v8f XY = __builtin_amdgcn_wmma_f32_16x16x4_f32(false, Ax, false, Bx,
                                                       (short)0, c0, false, false);
        v8f PP = __builtin_amdgcn_wmma_f32_16x16x4_f32(false, Ap, false, Bp,
                                                       (short)0, c0, false, false);

        const float y2 = xj0*xj0 + xj1*xj1 + xj2*xj2;
#pragma unroll
        for (int r = 0; r < 8; ++r) {
            float d2 = fmaf(-2.0f, XY[r], x2r[r] + y2);      // |x_i - x_j|^2
            float K  = __builtin_amdgcn_exp2f(C2EXP * d2);   // exp(-gamma*d2)
            float W  = K * PP[r];
            rs[r] += W;
            dxa[r] = fmaf(K, pj0, dxa[r]);
            dxb[r] = fmaf(K, pj1, dxb[r]);
            dxc[r] = fmaf(K, pj2, dxc[r]);
            wxa[r] = fmaf(W, xj0, wxa[r]);
            wxb[r] = fmaf(W, xj1, wxb[r]);
            wxc[r] = fmaf(W, xj2, wxc[r]);
        }
    }

    // reduce across the 16 lanes holding each row (xor masks < 16 stay in-half)
#pragma unroll
    for (int r = 0; r < 8; ++r) {
#pragma unroll
        for (int m = 8; m >= 1; m >>= 1) {
            rs[r]  += __shfl_xor(rs[r],  m, 32);
            dxa[r] += __shfl_xor(dxa[r], m, 32);
            dxb[r] += __shfl_xor(dxb[r], m, 32);
            dxc[r] += __shfl_xor(dxc[r], m, 32);
            wxa[r] += __shfl_xor(wxa[r], m, 32);
            wxb[r] += __shfl_xor(wxb[r], m, 32);
            wxc[r] += __shfl_xor(wxc[r], m, 32);
        }
    }

    if (l15 == 0) {
        const int rb = hi ? 8 : 0;
#pragma unroll
        for (int r = 0; r < 8; ++r) {
            partial[wave][rb + r][0] = rs[r];
            partial[wave][rb + r][1] = dxa[r];
            partial[wave][rb + r][2] = dxb[r];
            partial[wave][rb + r][3] = dxc[r];
            partial[wave][rb + r][4] = wxa[r];
            partial[wave][rb + r][5] = wxb[r];
            partial[wave][rb + r][6] = wxc[r];
        }
    }
    __syncthreads();

    if (threadIdx.x < 16) {
        const int row = threadIdx.x;
        float s0 = 0, s1 = 0, s2 = 0, s3 = 0, s4 = 0, s5 = 0, s6 = 0;
#pragma unroll
        for (int w = 0; w < 8; ++w) {
            s0 += partial[w][row][0];
            s1 += partial[w][row][1];
            s2 += partial[w][row][2];
            s3 += partial[w][row][3];
            s4 += partial[w][row][4];
            s5 += partial[w][row][5];
            s6 += partial[w][row][6];
        }
        const int i = i0 + row;
        const float xi0 = x[3*i], xi1 = x[3*i+1], xi2 = x[3*i+2];
        const float g2 = 2.0f * GAMMA_F;
        // dmom = 2*gamma*(x_i * rowsum - Wx)
        out[3*i]     = g2 * (xi0 * s0 - s4);
        out[3*i + 1] = g2 * (xi1 * s0 - s5);
        out[3*i + 2] = g2 * (xi2 * s0 - s6);
        // dx = K @ mom
        out[3*N + 3*i]     = s1;
        out[3*N + 3*i + 1] = s2;
        out[3*N + 3*i + 2] = s3;
    }
}

extern "C" void kernel_launch(void* const* d_in, const int* in_sizes, int n_in,
                              void* d_out, int out_size, void* d_ws, size_t ws_size,
                              hipStream_t stream) {
    const float* mom = (const float*)d_in[0];       // [1, N, 3] float32
    const float* cps = (const float*)d_in[1];       // [1, N, 3] float32
    float* out = (float*)d_out;                     // dmom (N*3) then dx (N*3)
    const int N = in_sizes[0] / 3;
    const int blocks = N >> 4;                      // 16 rows per block
    lddmm_evolve_kernel<<<blocks, 256, 0, stream>>>(mom, cps, out, N);
}